// HeirarchicalHashEmbedderNative_19705309954572
// MI455X (gfx1250) — compile-verified
//
#include <hip/hip_runtime.h>
#include <stdint.h>

// ---------------------------------------------------------------------------
// Multiresolution hash-grid encode (Instant-NGP style) for MI455X / gfx1250.
// Gather-bound kernel: 256M independent 8B gathers from a 512MB table.
// Design: wave32, 32-bit voffset + uniform base gathers, 16 loads in flight
// per thread (2 levels/iter), TDM async coord staging into LDS, b128 stores.
// ---------------------------------------------------------------------------

#define NLEVELS 16
#define TMASK   ((1u << 19) - 1u)

typedef __attribute__((ext_vector_type(4))) unsigned int tdm_g0_t;
typedef __attribute__((ext_vector_type(8))) int          tdm_g1_t;
typedef __attribute__((ext_vector_type(4))) int          tdm_g2_t;

// floor(16 * 1.5^l) for l in 0..15 (exact in f32)
__device__ __constant__ float kRes[NLEVELS] = {
    16.f, 24.f, 36.f, 54.f, 81.f, 121.f, 182.f, 273.f,
    410.f, 615.f, 922.f, 1383.f, 2075.f, 3113.f, 4670.f, 7006.f
};

__device__ __forceinline__ void level_setup(
    float res, uint32_t lvlBase, float cx, float cy, float cz,
    uint32_t off[8], float w[8])
{
    float sx = cx * res, sy = cy * res, sz = cz * res;
    float fx = floorf(sx), fy = floorf(sy), fz = floorf(sz);
    float tx = sx - fx, ty = sy - fy, tz = sz - fz;
    uint32_t x0 = (uint32_t)fx, y0 = (uint32_t)fy, z0 = (uint32_t)fz;
    // incremental corner hashes: (v+1)*prime = v*prime + prime (mod 2^32)
    uint32_t hx[2] = { x0, x0 + 1u };
    uint32_t hyb   = y0 * 2654435761u;
    uint32_t hy[2] = { hyb, hyb + 2654435761u };
    uint32_t hzb   = z0 * 805459861u;
    uint32_t hz[2] = { hzb, hzb + 805459861u };
    float wx[2] = { 1.0f - tx, tx };
    float wy[2] = { 1.0f - ty, ty };
    float wz[2] = { 1.0f - tz, tz };
#pragma unroll
    for (int c = 0; c < 8; ++c) {
        uint32_t h = (hx[c & 1] ^ hy[(c >> 1) & 1] ^ hz[(c >> 2) & 1]) & TMASK;
        off[c] = lvlBase | (h << 3);              // byte offset, fits in u32 (table = 2^29 B)
        w[c]   = wx[c & 1] * wy[(c >> 1) & 1] * wz[(c >> 2) & 1];
    }
}

__global__ __launch_bounds__(256)
void hashgrid_encode_kernel(const float* __restrict__ coords,
                            const float* __restrict__ tables,
                            float* __restrict__ out,
                            int n)
{
    __shared__ float sC[256 * 3];

    const int tid = threadIdx.x;
    const int gid = blockIdx.x * 256 + tid;

    // ---- TDM: async-stage this block's 256x3 f32 coord tile into LDS ------
    if (tid < 32) {  // wave 0 only issues the DMA (TDM ignores EXEC; one op/block)
        const uint32_t totalElems = (uint32_t)n * 3u;
        const uint32_t startElem  = (uint32_t)blockIdx.x * 768u;
        uint32_t nE = totalElems - startElem;
        if (nE > 768u) nE = 768u;                 // clip last (partial) tile

        const unsigned long long ga =
            (unsigned long long)(uintptr_t)coords + (unsigned long long)startElem * 4ull;
        const uint32_t ldsAddr = (uint32_t)(uintptr_t)&sC[0];

        // D# group0 (ISA 8.3): count=1 | lds_addr | global_addr | type=2
        tdm_g0_t g0;
        g0.x = 1u;                                          // count=1, user mode
        g0.y = ldsAddr;                                     // lds_addr[31:0]
        g0.z = (uint32_t)(ga & 0xFFFFFFFFull);              // global_addr[31:0]
        g0.w = (uint32_t)((ga >> 32) & 0x1FFFFFFull)        // global_addr[56:32]
             | 0x80000000u;                                 // type=2 ("image")

        // D# group1 (ISA 8.4): 1-D tile of nE 4-byte elements
        tdm_g1_t g1;
        g1[0] = (int)(2u << 16);                            // data_size=2 (4B), wg_mask=0
        g1[1] = (int)(nE << 16);                            // tensor_dim0[15:0]
        g1[2] = (int)((nE >> 16) | (1u << 16));             // tensor_dim0[31:16] | tensor_dim1=1
        g1[3] = (int)(nE << 16);                            // tile_dim0 = nE
        g1[4] = 0;                                          // tile_dim1=0, tile_dim2=0 (unused)
        g1[5] = (int)nE;                                    // tensor_dim0_stride lo
        g1[6] = 0;
        g1[7] = 0;

        tdm_g2_t gz = { 0, 0, 0, 0 };                       // groups 2/3: dims 3/4 unused
#if __clang_major__ >= 23
        tdm_g1_t gz8 = { 0, 0, 0, 0, 0, 0, 0, 0 };
        __builtin_amdgcn_tensor_load_to_lds(g0, g1, gz, gz, gz8, 0);
#else
        __builtin_amdgcn_tensor_load_to_lds(g0, g1, gz, gz, 0);
#endif
        __builtin_amdgcn_s_wait_tensorcnt((short)0);
    }
    __syncthreads();

    if (gid >= n) return;

    // coords from LDS (ds_load)
    const float cx = sC[tid * 3 + 0];
    const float cy = sC[tid * 3 + 1];
    const float cz = sC[tid * 3 + 2];

    // spatial partition: cell = clip(floor(c*2), 0, 1); flat = x*4 + y*2 + z
    int ix = (int)floorf(cx * 2.0f); ix = ix < 0 ? 0 : (ix > 1 ? 1 : ix);
    int iy = (int)floorf(cy * 2.0f); iy = iy < 0 ? 0 : (iy > 1 ? 1 : iy);
    int iz = (int)floorf(cz * 2.0f); iz = iz < 0 ? 0 : (iz > 1 ? 1 : iz);
    const uint32_t partOff = (uint32_t)(ix * 4 + iy * 2 + iz) << 26;  // 64MB / partition

    const char* __restrict__ tb = (const char*)tables;  // uniform SGPR base
    float* __restrict__ orow = out + (size_t)gid * 32;

#pragma unroll
    for (int l = 0; l < NLEVELS; l += 2) {
        uint32_t off0[8], off1[8];
        float w0[8], w1[8];
        level_setup(kRes[l],     partOff | ((uint32_t)l       << 22), cx, cy, cz, off0, w0);
        level_setup(kRes[l + 1], partOff | ((uint32_t)(l + 1) << 22), cx, cy, cz, off1, w1);

        // 16 independent global_load_b64 gathers in flight
        float2 e0[8], e1[8];
#pragma unroll
        for (int c = 0; c < 8; ++c) e0[c] = *(const float2*)(tb + off0[c]);
#pragma unroll
        for (int c = 0; c < 8; ++c) e1[c] = *(const float2*)(tb + off1[c]);

        float a0 = 0.f, a1 = 0.f, b0 = 0.f, b1 = 0.f;
#pragma unroll
        for (int c = 0; c < 8; ++c) {
            a0 = fmaf(w0[c], e0[c].x, a0);
            a1 = fmaf(w0[c], e0[c].y, a1);
        }
#pragma unroll
        for (int c = 0; c < 8; ++c) {
            b0 = fmaf(w1[c], e1[c].x, b0);
            b1 = fmaf(w1[c], e1[c].y, b1);
        }

        float4 r; r.x = a0; r.y = a1; r.z = b0; r.w = b1;
        *(float4*)(orow + l * 2) = r;                       // 16B-aligned b128 store
    }
}

extern "C" void kernel_launch(void* const* d_in, const int* in_sizes, int n_in,
                              void* d_out, int out_size, void* d_ws, size_t ws_size,
                              hipStream_t stream)
{
    const float* coords = (const float*)d_in[0];   // (N, 3) f32
    const float* tables = (const float*)d_in[1];   // (8, 16, 2^19, 2) f32
    float* out = (float*)d_out;                    // (N, 32) f32

    const int n = in_sizes[0] / 3;
    const int blocks = (n + 255) / 256;
    hashgrid_encode_kernel<<<blocks, 256, 0, stream>>>(coords, tables, out, n);
}